// DeformConv3D_5669356831603
// MI455X (gfx1250) — compile-verified
//
#include <hip/hip_runtime.h>
#include <hip/hip_bf16.h>

// ---------------------------------------------------------------------------
// Deformable Conv3D (B=2, C=64, D=8, H=56, W=56, Cout=64, 3x3x3, s=1,p=1,d=1)
// Fused trilinear-sampling + im2col GEMM on V_WMMA_F32_16X16X4_F32.
// Weight slices staged per-tap into LDS by the Tensor Data Mover (TDM),
// double-buffered and overlapped with the gather phase.
// ---------------------------------------------------------------------------

#define NB    2
#define NC    64
#define ND    8
#define NH    56
#define NW    56
#define NCOUT 64
#define NK    27
#define HWP   (NH * NW)          // 3136
#define DHW   (ND * NH * NW)     // 25088
#define NP    DHW                // outputs per batch
#define NTILE (NP / 16)          // 1568 position tiles per batch

#define LSTR  68                 // padded LDS row stride (dwords): bank = 4r+c

typedef __attribute__((ext_vector_type(2))) float v2f;
typedef __attribute__((ext_vector_type(8))) float v8f;
typedef __attribute__((ext_vector_type(4))) unsigned int v4u;
typedef __attribute__((ext_vector_type(8))) int v8i;
typedef __attribute__((ext_vector_type(4))) int v4i;

__device__ __forceinline__ int iclamp(int v, int lo, int hi) {
    return v < lo ? lo : (v > hi ? hi : v);
}

// ---- TDM: DMA one contiguous 4096-dword weight slice into LDS, inserting
// ---- 4 pad dwords every 64 dwords (-> [64][68] conflict-free layout).
__device__ __forceinline__ void tdm_load_wslice(const float* gsrc,
                                                unsigned int lds_byte_off) {
    unsigned long long ga = (unsigned long long)(size_t)gsrc;
    v4u g0;
    g0[0] = 1u;                                   // count=1 user descriptor
    g0[1] = lds_byte_off;                         // lds_addr
    g0[2] = (unsigned int)(ga & 0xFFFFFFFFu);     // global_addr[31:0]
    g0[3] = (unsigned int)((ga >> 32) & 0x01FFFFFFu) | (2u << 30); // addr[56:32], type=2
    v8i g1;
    g1[0] = (int)((2u << 16)      // data_size = 4B
                | (1u << 20)      // pad_enable
                | (5u << 22)      // pad_interval: every 64 dwords
                | (3u << 25));    // pad_amount: 4 dwords
    g1[1] = (int)(4096u << 16);   // tensor_dim0[15:0] = 4096
    g1[2] = (int)(1u << 16);      // tensor_dim0 hi = 0, tensor_dim1 = 1
    g1[3] = (int)(4096u << 16);   // tensor_dim1 hi = 0, tile_dim0 = 4096
    g1[4] = 0;                    // tile_dim1/2 unused
    g1[5] = 4096;                 // tensor_dim0_stride
    g1[6] = 0;
    g1[7] = 0;
    v4i gz = {0, 0, 0, 0};
#if __has_include(<hip/amd_detail/amd_gfx1250_TDM.h>)
    v8i gz8 = {0, 0, 0, 0, 0, 0, 0, 0};
    __builtin_amdgcn_tensor_load_to_lds(g0, g1, gz, gz, gz8, 0);
#else
    __builtin_amdgcn_tensor_load_to_lds(g0, g1, gz, gz, 0);
#endif
}

// ---- repack weights [cout][cin][k] -> [k][cout][cin] (contiguous per tap) --
__global__ __launch_bounds__(256) void
repack_weights(const float* __restrict__ wgt, float* __restrict__ wrep) {
    int idx = blockIdx.x * 256 + threadIdx.x;     // 432 blocks * 256 = 110592
    int k   = idx / (NCOUT * NC);
    int rem = idx - k * (NCOUT * NC);             // rem = cout*64 + cin
    wrep[idx] = wgt[(size_t)rem * NK + k];
}

// ===========================================================================
// Main kernel, TDM weight staging variant.
// ===========================================================================
__global__ __launch_bounds__(128) void
dcn3d_wmma_tdm(const float* __restrict__ x,
               const float* __restrict__ offs,
               const float* __restrict__ wrep,   // [k][cout][cin]
               float* __restrict__ out)
{
    __shared__ __align__(16) float Wlds[2][NCOUT * LSTR]; // 2 x 17.4 KB
    __shared__ __align__(16) float Blds[16 * LSTR];       // cols^T [n][cin]
    __shared__ float smw[8][16];
    __shared__ int   smi[8][16];

    const int tid  = threadIdx.x;
    const int lane = tid & 31;
    const int wave = tid >> 5;
    const int half = lane >> 4;
    const int r    = lane & 15;

    const int blk = blockIdx.x;
    const int b   = blk / NTILE;
    const int p0  = (blk % NTILE) * 16;

    const float* xb = x + (size_t)b * NC * DHW;

    v8f acc0 = {}, acc1 = {};
    int cur = 0;

    // prologue: start DMA of tap 0's weight slice into buffer 0
    if (wave == 0)
        tdm_load_wslice(wrep, (unsigned int)(size_t)&Wlds[0][0]);

    for (int k = 0; k < NK; ++k) {
        __syncthreads();   // prior WMMA reads of Blds / Wlds[1-cur] complete

        // ---- kick off DMA of next tap's weights into the other buffer -----
        if (k < NK - 1 && wave == 0)
            tdm_load_wslice(wrep + (size_t)(k + 1) * NCOUT * NC,
                            (unsigned int)(size_t)&Wlds[cur ^ 1][0]);

        // ---- sampling metadata for the 16 output positions ----------------
        if (tid < 16) {
            int p   = p0 + tid;
            int dd  = p / HWP;
            int rem = p - dd * HWP;
            int hh  = rem / NW;
            int ww  = rem - hh * NW;
            int kd  = k / 9, kh = (k / 3) % 3, kw = k % 3;

            const float* ob = offs + ((size_t)b * (3 * NK) + (size_t)k * 3) * NP + p;
            float pd = ob[0]              + (float)(kd + dd - 1);
            float ph = ob[NP]             + (float)(kh + hh - 1);
            float pw = ob[2 * (size_t)NP] + (float)(kw + ww - 1);

            float d0f = floorf(pd), h0f = floorf(ph), w0f = floorf(pw);
            float fd = pd - d0f, fh = ph - h0f, fw = pw - w0f;
            int d0 = (int)d0f, h0 = (int)h0f, w0 = (int)w0f;

            #pragma unroll
            for (int c8 = 0; c8 < 8; ++c8) {
                int bd = (c8 >> 2) & 1, bh = (c8 >> 1) & 1, bw = c8 & 1;
                int di = d0 + bd, hi = h0 + bh, wi = w0 + bw;
                bool valid = (di >= 0) && (di < ND) &&
                             (hi >= 0) && (hi < NH) &&
                             (wi >= 0) && (wi < NW);
                float wd = bd ? fd : 1.0f - fd;
                float wh = bh ? fh : 1.0f - fh;
                float wv = bw ? fw : 1.0f - fw;
                int dic = iclamp(di, 0, ND - 1);
                int hic = iclamp(hi, 0, NH - 1);
                int wic = iclamp(wi, 0, NW - 1);
                smw[c8][tid] = valid ? (wd * wh * wv) : 0.0f;
                smi[c8][tid] = (dic * NH + hic) * NW + wic;
            }
        }
        __syncthreads();

        // ---- gather cols^T tile: thread (n = tid&15) covers 8 channels ----
        #pragma unroll
        for (int j = 0; j < 8; ++j) {
            int e = tid + j * 128;
            int n = e & 15;
            int c = e >> 4;
            const float* xc = xb + (size_t)c * DHW;
            float v = 0.0f;
            #pragma unroll
            for (int c8 = 0; c8 < 8; ++c8)
                v += smw[c8][n] * xc[smi[c8][n]];
            Blds[n * LSTR + c] = v;
        }

        // ---- current tap's weight DMA must be complete --------------------
        if (wave == 0) {
            if (k < NK - 1) __builtin_amdgcn_s_wait_tensorcnt(1);
            else            __builtin_amdgcn_s_wait_tensorcnt(0);
        }
        __syncthreads();   // Blds + Wlds[cur] visible to every wave

        // ---- WMMA: 16 K=4 steps, dual accumulators break the RAW chain ----
        const float* Wc = &Wlds[cur][0];
        #pragma unroll
        for (int kk = 0; kk < 16; kk += 2) {
            int c0 = kk * 4 + half * 2;
            v2f a0 = *(const v2f*)&Wc[(wave * 16 + r) * LSTR + c0];
            v2f b0 = *(const v2f*)&Blds[r * LSTR + c0];
            v2f a1 = *(const v2f*)&Wc[(wave * 16 + r) * LSTR + c0 + 4];
            v2f b1 = *(const v2f*)&Blds[r * LSTR + c0 + 4];
            acc0 = __builtin_amdgcn_wmma_f32_16x16x4_f32(
                false, a0, false, b0, (short)0, acc0, false, false);
            acc1 = __builtin_amdgcn_wmma_f32_16x16x4_f32(
                false, a1, false, b1, (short)0, acc1, false, false);
        }
        cur ^= 1;
    }

    v8f acc = acc0 + acc1;
    float* ob = out + (size_t)b * NCOUT * NP + p0;
    #pragma unroll
    for (int i = 0; i < 8; ++i) {
        int cout = wave * 16 + i + 8 * half;
        ob[(size_t)cout * NP + r] = acc[i];
    }
}

// ===========================================================================
// Fallback (no workspace): stage weights with plain loads, same math core.
// ===========================================================================
__global__ __launch_bounds__(128) void
dcn3d_wmma_stage(const float* __restrict__ x,
                 const float* __restrict__ offs,
                 const float* __restrict__ wgt,
                 float* __restrict__ out)
{
    __shared__ __align__(16) float Wlds[NCOUT * LSTR];
    __shared__ __align__(16) float Blds[16 * LSTR];
    __shared__ float smw[8][16];
    __shared__ int   smi[8][16];

    const int tid  = threadIdx.x;
    const int lane = tid & 31;
    const int wave = tid >> 5;
    const int half = lane >> 4;
    const int r    = lane & 15;

    const int blk = blockIdx.x;
    const int b   = blk / NTILE;
    const int p0  = (blk % NTILE) * 16;

    const float* xb = x + (size_t)b * NC * DHW;

    v8f acc0 = {}, acc1 = {};

    for (int k = 0; k < NK; ++k) {
        __syncthreads();

        #pragma unroll
        for (int j = 0; j < 32; ++j) {
            int e = tid + j * 128;                       // e = cout*64 + cin
            Wlds[e + 4 * (e >> 6)] = wgt[(size_t)e * NK + k];
        }
        if (k < NK - 1)
            __builtin_prefetch(&wgt[(size_t)(tid * 32) * NK + k + 1], 0, 0);

        if (tid < 16) {
            int p   = p0 + tid;
            int dd  = p / HWP;
            int rem = p - dd * HWP;
            int hh  = rem / NW;
            int ww  = rem - hh * NW;
            int kd  = k / 9, kh = (k / 3) % 3, kw = k % 3;

            const float* ob = offs + ((size_t)b * (3 * NK) + (size_t)k * 3) * NP + p;
            float pd = ob[0]              + (float)(kd + dd - 1);
            float ph = ob[NP]             + (float)(kh + hh - 1);
            float pw = ob[2 * (size_t)NP] + (float)(kw + ww - 1);

            float d0f = floorf(pd), h0f = floorf(ph), w0f = floorf(pw);
            float fd = pd - d0f, fh = ph - h0f, fw = pw - w0f;
            int d0 = (int)d0f, h0 = (int)h0f, w0 = (int)w0f;

            #pragma unroll
            for (int c8 = 0; c8 < 8; ++c8) {
                int bd = (c8 >> 2) & 1, bh = (c8 >> 1) & 1, bw = c8 & 1;
                int di = d0 + bd, hi = h0 + bh, wi = w0 + bw;
                bool valid = (di >= 0) && (di < ND) &&
                             (hi >= 0) && (hi < NH) &&
                             (wi >= 0) && (wi < NW);
                float wd = bd ? fd : 1.0f - fd;
                float wh = bh ? fh : 1.0f - fh;
                float wv = bw ? fw : 1.0f - fw;
                smw[c8][tid] = valid ? (wd * wh * wv) : 0.0f;
                smi[c8][tid] = (iclamp(di, 0, ND - 1) * NH +
                                iclamp(hi, 0, NH - 1)) * NW +
                               iclamp(wi, 0, NW - 1);
            }
        }
        __syncthreads();

        #pragma unroll
        for (int j = 0; j < 8; ++j) {
            int e = tid + j * 128;
            int n = e & 15;
            int c = e >> 4;
            const float* xc = xb + (size_t)c * DHW;
            float v = 0.0f;
            #pragma unroll
            for (int c8 = 0; c8 < 8; ++c8)
                v += smw[c8][n] * xc[smi[c8][n]];
            Blds[n * LSTR + c] = v;
        }
        __syncthreads();

        #pragma unroll
        for (int kk = 0; kk < 16; kk += 2) {
            int c0 = kk * 4 + half * 2;
            v2f a0 = *(const v2f*)&Wlds[(wave * 16 + r) * LSTR + c0];
            v2f b0 = *(const v2f*)&Blds[r * LSTR + c0];
            v2f a1 = *(const v2f*)&Wlds[(wave * 16 + r) * LSTR + c0 + 4];
            v2f b1 = *(const v2f*)&Blds[r * LSTR + c0 + 4];
            acc0 = __builtin_amdgcn_wmma_f32_16x16x4_f32(
                false, a0, false, b0, (short)0, acc0, false, false);
            acc1 = __builtin_amdgcn_wmma_f32_16x16x4_f32(
                false, a1, false, b1, (short)0, acc1, false, false);
        }
    }

    v8f acc = acc0 + acc1;
    float* ob = out + (size_t)b * NCOUT * NP + p0;
    #pragma unroll
    for (int i = 0; i < 8; ++i) {
        int cout = wave * 16 + i + 8 * half;
        ob[(size_t)cout * NP + r] = acc[i];
    }
}

extern "C" void kernel_launch(void* const* d_in, const int* in_sizes, int n_in,
                              void* d_out, int out_size, void* d_ws, size_t ws_size,
                              hipStream_t stream) {
    (void)in_sizes; (void)n_in; (void)out_size;
    const float* x    = (const float*)d_in[0];
    const float* offs = (const float*)d_in[1];
    const float* wgt  = (const float*)d_in[2];
    float* out        = (float*)d_out;

    const size_t wrep_bytes = (size_t)NK * NCOUT * NC * sizeof(float); // 442 KB

    dim3 grid(NB * NTILE);   // 3136 workgroups
    dim3 block(128);         // 4 wave32s

    if (ws_size >= wrep_bytes) {
        float* wrep = (float*)d_ws;
        repack_weights<<<dim3((NK * NCOUT * NC) / 256), dim3(256), 0, stream>>>(wgt, wrep);
        dcn3d_wmma_tdm<<<grid, block, 0, stream>>>(x, offs, wrep, out);
    } else {
        dcn3d_wmma_stage<<<grid, block, 0, stream>>>(x, offs, wgt, out);
    }
}